// ColPali2Loss_78889959292952
// MI455X (gfx1250) — compile-verified
//
#include <hip/hip_runtime.h>
#include <math.h>

// ---- problem constants (from reference) ----
#define BSZ   64      // batch
#define NQ    32      // query tokens
#define NS    1024    // doc tokens
#define DIM   128     // embedding dim
#define ALPHA 0.5f

#define LPAD  132     // padded LDS row stride in floats (128+4) -> conflict-free
#define TILES (NS / 16)
#define NWAVE 2       // waves per block (64 threads)

typedef __attribute__((ext_vector_type(2))) float v2f;
typedef __attribute__((ext_vector_type(8))) float v8f;
typedef __attribute__((ext_vector_type(4))) unsigned int v4u;
typedef __attribute__((ext_vector_type(8))) int v8i;
typedef __attribute__((ext_vector_type(4))) int v4i;

// ---------------------------------------------------------------------
// TDM: DMA one 16x128 f32 tile (row stride DIM in memory) into LDS with
// a 4-DWORD pad after every 128 DWORDs -> LDS row stride = 132 floats.
// D# group0: count=1 | lds_addr | global_addr(57b) | type=2
// D# group1: data_size=4B, pad_enable, pad_interval=128dw, pad_amount=4dw,
//            tensor_dim0=128, tensor_dim1=16, tile_dim0=128, tile_dim1=16,
//            tensor_dim0_stride=128
// ---------------------------------------------------------------------
__device__ __forceinline__ void tdm_load_tile(const float* gsrc, unsigned lds_byte_off) {
    unsigned long long ga = (unsigned long long)(uintptr_t)gsrc;
    v4u g0;
    g0.x = 0x1u;                                        // count=1, user mode
    g0.y = lds_byte_off;                                // lds_addr (bytes)
    g0.z = (unsigned)(ga & 0xffffffffu);                // global_addr[31:0]
    g0.w = (unsigned)((ga >> 32) & 0x1ffffffu)          // global_addr[56:32]
         | (2u << 30);                                  // type = 2 ("image")
    v8i g1;
    g1[0] = (int)((2u << 16)        // data_size = 4 bytes
                | (1u << 20)        // pad_enable
                | (6u << 22)        // pad_interval: 128 DWORDs
                | (3u << 25));      // pad_amount:   4 DWORDs
    g1[1] = (int)(DIM << 16);       // tensor_dim0 = 128 (bits 79:48 low half)
    g1[2] = (int)(16u << 16);       // tensor_dim1 = 16  (bits 111:80 low half)
    g1[3] = (int)(DIM << 16);       // tile_dim0   = 128 (bits 127:112)
    g1[4] = 16;                     // tile_dim1   = 16, tile_dim2 = 0
    g1[5] = DIM;                    // tensor_dim0_stride = 128 (low 32)
    g1[6] = 0;                      // stride hi / dim1_stride lo
    g1[7] = 0;
    v4i z4 = {0, 0, 0, 0};          // groups 2/3 unused (2D tensor)
    v8i z8 = {0, 0, 0, 0, 0, 0, 0, 0};
    // 6-arg form on this toolchain: (g0, g1, g2, g3, extra_group, cpol)
    __builtin_amdgcn_tensor_load_to_lds(g0, g1, z4, z4, z8, 0);
}

__device__ __forceinline__ unsigned lds_off(const void* p) {
    // generic LDS address: addr[31:0] is the LDS byte offset (ISA aperture rule)
    return (unsigned)(uintptr_t)p;
}

// =====================================================================
// Kernel 1: MaxSim scores.  One 64-thread block (2 waves) per (b,c) pair.
//   scores[b][c] = sum_n max_s  dot(q[b,n,:], d[c,s,:])
// A-matrix = 16 doc tokens (M) x K, B-matrix = K x 16 query tokens (N).
// C layout (f32 16x16): lane holds column N = lane&15, rows M = vgpr + 8*(lane>>4).
// =====================================================================
__global__ __launch_bounds__(64) void
mv_scores_kernel(const float* __restrict__ qm,   // (B, NQ, DIM)
                 const float* __restrict__ dm,   // (B, NS, DIM)
                 float* __restrict__ scores)     // (B, B)
{
    __shared__ float qlds[NQ * LPAD];                 // 16.9 KB
    __shared__ float dlds[NWAVE][2][16 * LPAD];       // 33.8 KB (double-buffered)
    __shared__ float wmax[NWAVE][NQ];

    const int b    = blockIdx.x;
    const int c    = blockIdx.y;
    const int tid  = threadIdx.x;
    const int wave = tid >> 5;
    const int lane = tid & 31;

    // ---- stage q[b] (32 x 128) into LDS, padded rows ----
    const float* qbase = qm + (size_t)b * NQ * DIM;
    for (int i = tid; i < (NQ * DIM) / 4; i += 64) {
        const int r = (i * 4) / DIM, col = (i * 4) % DIM;
        const float4 v = *(const float4*)(qbase + r * DIM + col);
        *(float4*)(&qlds[r * LPAD + col]) = v;
    }
    __syncthreads();

    const float* dbase = dm + (size_t)c * NS * DIM;

    // A/B fragment addressing (32-bit 16x4 layout):
    //   VGPR0: lanes 0-15 -> K=kb+0, lanes 16-31 -> K=kb+2 ; VGPR1: K=kb+1 / kb+3
    const int row   = lane & 15;          // doc token (A) / query token (B)
    const int khalf = (lane >> 4) << 1;   // 0 or 2

    // ---- hoist ALL query B-fragments into registers (q[b] is invariant
    //      across the 64 doc tiles): 2 halves x 32 k-steps x v2f = 128 VGPRs.
    //      Occupancy is LDS-limited, so these registers are free.
    v2f qf0[DIM / 4], qf1[DIM / 4];
    {
        const float* b0 = &qlds[(row) * LPAD + khalf];          // query tokens 0..15
        const float* b1 = &qlds[(16 + row) * LPAD + khalf];     // query tokens 16..31
        #pragma unroll
        for (int ks = 0; ks < DIM / 4; ++ks) {
            qf0[ks] = *(const v2f*)(b0 + 4 * ks);
            qf1[ks] = *(const v2f*)(b1 + 4 * ks);
        }
    }

    float runmax0 = -INFINITY;            // query tokens 0..15  (token = lane&15)
    float runmax1 = -INFINITY;            // query tokens 16..31

    // prime the pipeline: DMA first tile for this wave into buffer 0
    tdm_load_tile(dbase + (size_t)wave * 16 * DIM, lds_off(&dlds[wave][0][0]));

    int buf = 0;
    for (int t = wave; t < TILES; t += NWAVE) {
        // wait for the DMA that fills the current buffer
        __builtin_amdgcn_s_wait_tensorcnt(0);
        asm volatile("" ::: "memory");

        // kick off DMA for the next tile into the other buffer (overlaps compute)
        const int nt = t + NWAVE;
        if (nt < TILES)
            tdm_load_tile(dbase + (size_t)nt * 16 * DIM,
                          lds_off(&dlds[wave][buf ^ 1][0]));

        const float* aptr = &dlds[wave][buf][0] + row * LPAD + khalf;

        v8f acc0 = {}, acc1 = {};
        #pragma unroll
        for (int ks = 0; ks < DIM / 4; ++ks) {
            v2f a = *(const v2f*)(aptr + 4 * ks);
            acc0 = __builtin_amdgcn_wmma_f32_16x16x4_f32(
                       false, a, false, qf0[ks], (short)0, acc0, false, false);
            acc1 = __builtin_amdgcn_wmma_f32_16x16x4_f32(
                       false, a, false, qf1[ks], (short)0, acc1, false, false);
        }

        // max over this lane's 8 doc rows, then the other M-half (lanes xor 16)
        float m0 = fmaxf(fmaxf(fmaxf(acc0[0], acc0[1]), fmaxf(acc0[2], acc0[3])),
                         fmaxf(fmaxf(acc0[4], acc0[5]), fmaxf(acc0[6], acc0[7])));
        float m1 = fmaxf(fmaxf(fmaxf(acc1[0], acc1[1]), fmaxf(acc1[2], acc1[3])),
                         fmaxf(fmaxf(acc1[4], acc1[5]), fmaxf(acc1[6], acc1[7])));
        m0 = fmaxf(m0, __shfl_xor(m0, 16, 32));
        m1 = fmaxf(m1, __shfl_xor(m1, 16, 32));
        runmax0 = fmaxf(runmax0, m0);
        runmax1 = fmaxf(runmax1, m1);

        buf ^= 1;
    }

    if (lane < 16) {
        wmax[wave][lane]      = runmax0;
        wmax[wave][16 + lane] = runmax1;
    }
    __syncthreads();

    if (tid < NQ) {                        // wave 0 only
        float m = wmax[0][tid];
        #pragma unroll
        for (int w = 1; w < NWAVE; ++w) m = fmaxf(m, wmax[w][tid]);
        // sum the 32 per-query-token maxima across the wave
        #pragma unroll
        for (int off = 16; off > 0; off >>= 1) m += __shfl_xor(m, off, 32);
        if (tid == 0) scores[b * BSZ + c] = m;
    }
}

// =====================================================================
// Kernel 2: sv log-softmax loss + mv softplus margin loss + total.
// Single block of 64 threads, one row per thread.
// =====================================================================
__device__ __forceinline__ float softplus_stable(float x) {
    return fmaxf(x, 0.0f) + log1pf(expf(-fabsf(x)));
}

__global__ __launch_bounds__(64) void
finalize_kernel(const float* __restrict__ qs,      // (B, DIM)
                const float* __restrict__ ds,      // (B, DIM)
                const float* __restrict__ scores,  // (B, B) from kernel 1
                float* __restrict__ out)           // [sv, mv, total]
{
    __shared__ float dl[BSZ * DIM];    // 32 KB doc_single cache
    __shared__ float red_sv[BSZ];
    __shared__ float red_mv[BSZ];

    const int t = threadIdx.x;

    for (int i = t; i < (BSZ * DIM) / 4; i += 64)
        ((float4*)dl)[i] = ((const float4*)ds)[i];
    __syncthreads();

    // cache this row's query vector in registers
    float q[DIM];
    #pragma unroll
    for (int k = 0; k < DIM; ++k) q[k] = qs[t * DIM + k];

    // ---- single-vector loss row t: lse(row) - diag ----
    float rmax = -INFINITY, diag = 0.0f;
    for (int cc = 0; cc < BSZ; ++cc) {
        const float* dr = &dl[cc * DIM];
        float s = 0.0f;
        #pragma unroll 8
        for (int k = 0; k < DIM; ++k) s = fmaf(q[k], dr[k], s);
        if (cc == t) diag = s;
        rmax = fmaxf(rmax, s);
    }
    float esum = 0.0f;
    for (int cc = 0; cc < BSZ; ++cc) {
        const float* dr = &dl[cc * DIM];
        float s = 0.0f;
        #pragma unroll 8
        for (int k = 0; k < DIM; ++k) s = fmaf(q[k], dr[k], s);
        esum += expf(s - rmax);
    }
    red_sv[t] = (rmax + logf(esum)) - diag;

    // ---- multi-vector margin loss row t ----
    const float* srow = scores + t * BSZ;
    float pos = srow[t];
    float neg = -INFINITY;
    for (int cc = 0; cc < BSZ; ++cc) {
        float v = srow[cc] - ((cc == t) ? 1.0e6f : 0.0f);
        neg = fmaxf(neg, v);
    }
    red_mv[t] = softplus_stable(neg - pos);
    __syncthreads();

    if (t == 0) {
        float sv = 0.0f, mv = 0.0f;
        for (int i = 0; i < BSZ; ++i) { sv += red_sv[i]; mv += red_mv[i]; }
        sv /= (float)BSZ;
        mv /= (float)BSZ;
        out[0] = sv;
        out[1] = mv;
        out[2] = ALPHA * sv + (1.0f - ALPHA) * mv;
    }
}

// =====================================================================
extern "C" void kernel_launch(void* const* d_in, const int* in_sizes, int n_in,
                              void* d_out, int out_size, void* d_ws, size_t ws_size,
                              hipStream_t stream) {
    const float* query_single = (const float*)d_in[0];   // (64,128)
    const float* doc_single   = (const float*)d_in[1];   // (64,128)
    const float* query_multi  = (const float*)d_in[2];   // (64,32,128)
    const float* doc_multi    = (const float*)d_in[3];   // (64,1024,128)
    float* out    = (float*)d_out;
    float* scores = (float*)d_ws;                        // 64*64 floats = 16 KB

    dim3 grid(BSZ, BSZ);
    mv_scores_kernel<<<grid, 64, 0, stream>>>(query_multi, doc_multi, scores);
    finalize_kernel<<<1, 64, 0, stream>>>(query_single, doc_single, scores, out);
}